// hyperedge_attn_9543417332145
// MI455X (gfx1250) — compile-verified
//
#include <hip/hip_runtime.h>
#include <math.h>

#define B_  16
#define S_  1024
#define E_  64
#define K_  32
#define F_  128
#define HD_ 64

typedef __attribute__((ext_vector_type(2))) float v2f;
typedef __attribute__((ext_vector_type(8))) float v8f;

// D = A(16x4 f32) * B(4x16 f32) + C(16x16 f32), wave32 WMMA
__device__ __forceinline__ v8f wmma4(v2f a, v2f b, v8f c) {
  return __builtin_amdgcn_wmma_f32_16x16x4_f32(false, a, false, b, (short)0, c,
                                               false, false);
}

__device__ __forceinline__ float leaky(float v) { return v > 0.f ? v : 0.01f * v; }
__device__ __forceinline__ float elu(float v)   { return v > 0.f ? v : (expf(v) - 1.f); }

// ---------------------------------------------------------------------------
// 0) zero-fill the sparse scatter output (268 MB) with wide stores
// ---------------------------------------------------------------------------
__global__ void zero4(float4* __restrict__ p, size_t n4) {
  size_t i = (size_t)blockIdx.x * blockDim.x + threadIdx.x;
  if (i < n4) p[i] = make_float4(0.f, 0.f, 0.f, 0.f);
}

// ---------------------------------------------------------------------------
// 1) industry prep: seq = x@W1_g + b1, resid = x@Wr_g + br for all B*S rows.
//    grid = (B*S)/16 blocks, 128 threads (4 waves = 4 N-tiles of 16 cols).
// ---------------------------------------------------------------------------
__global__ __launch_bounds__(128) void prep_industry(
    const float* __restrict__ x, const float* __restrict__ W1,
    const float* __restrict__ b1, const float* __restrict__ Wr,
    const float* __restrict__ br, float* __restrict__ seq,
    float* __restrict__ resid) {
  const int row0 = blockIdx.x * 16;
  const int wave = threadIdx.x >> 5;
  const int lane = threadIdx.x & 31;
  const int hi   = lane >> 4;
  const int ln   = lane & 15;
  const int n0   = wave * 16;
  const int n    = n0 + ln;

  v8f cs = {}; v8f cr = {};
  const float* xrow = x + (size_t)(row0 + ln) * F_;
#pragma unroll
  for (int kk = 0; kk < F_; kk += 4) {
    const int k = kk + 2 * hi;
    v2f a = *(const v2f*)(xrow + k);                    // x[m][k], x[m][k+1]
    v2f bw; bw.x = W1[(size_t)k * HD_ + n]; bw.y = W1[(size_t)(k + 1) * HD_ + n];
    v2f bv; bv.x = Wr[(size_t)k * HD_ + n]; bv.y = Wr[(size_t)(k + 1) * HD_ + n];
    cs = wmma4(a, bw, cs);
    cr = wmma4(a, bv, cr);
  }
  const float bb1 = b1[n], bbr = br[n];
#pragma unroll
  for (int r = 0; r < 8; ++r) {
    const int m = row0 + r + 8 * hi;
    seq  [(size_t)m * HD_ + n] = cs[r] + bb1;
    resid[(size_t)m * HD_ + n] = cr[r] + bbr;
  }
}

// ---------------------------------------------------------------------------
// 2) f1/f2 = seq.a1 + ba1, seq.a2 + ba2 per row (trivial, 1 thread / row)
// ---------------------------------------------------------------------------
__global__ void f12_g(const float* __restrict__ seq, const float* __restrict__ a1,
                      const float* __restrict__ ba1, const float* __restrict__ a2,
                      const float* __restrict__ ba2, float* __restrict__ f1,
                      float* __restrict__ f2, int nrows) {
  int r = blockIdx.x * blockDim.x + threadIdx.x;
  if (r >= nrows) return;
  float s1 = ba1[0], s2 = ba2[0];
  const float* p = seq + (size_t)r * HD_;
#pragma unroll 8
  for (int k = 0; k < HD_; ++k) { float v = p[k]; s1 += v * a1[k]; s2 += v * a2[k]; }
  f1[r] = s1; f2[r] = s2;
}

// ---------------------------------------------------------------------------
// 3) industry attention: one block per (b, 16-row tile); masked softmax over
//    S=1024 cols, then P@seq via WMMA in two 512-col LDS chunks.
// ---------------------------------------------------------------------------
#define CH 512
__global__ __launch_bounds__(128) void industry_attn(
    const float* __restrict__ seq, const float* __restrict__ resid,
    const float* __restrict__ f1, const float* __restrict__ f2,
    const float* __restrict__ adj, float* __restrict__ out_ind) {
  const int b    = blockIdx.x / (S_ / 16);
  const int tile = blockIdx.x % (S_ / 16);
  const int row0 = tile * 16;

  __shared__ float P[16][CH];          // unnormalized softmax chunk
  __shared__ float rowmax[16], rowsum[16];
  __shared__ float red[128];

  const int i  = threadIdx.x & 15;     // row within tile (fill phase)
  const int jt = threadIdx.x >> 4;     // 0..7 column strider
  const float* f1b = f1 + (size_t)b * S_;
  const float* f2b = f2 + (size_t)b * S_;
  const float* adjrow = adj + (size_t)(row0 + i) * S_;
  const float fi = f2b[row0 + i];

  // pass 1: row maxima over all S columns
  float mx = -INFINITY;
  for (int j = jt; j < S_; j += 8) {
    float l = leaky(fi + f1b[j]) - 1e9f * (1.0f - adjrow[j]);
    mx = fmaxf(mx, l);
  }
  red[threadIdx.x] = mx;
  __syncthreads();
  if (jt == 0) {
    float m = red[i];
#pragma unroll
    for (int q = 1; q < 8; ++q) m = fmaxf(m, red[q * 16 + i]);
    rowmax[i] = m;
    rowsum[i] = 0.f;
  }
  __syncthreads();

  const int wave = threadIdx.x >> 5;
  const int lane = threadIdx.x & 31;
  const int hi   = lane >> 4;
  const int ln   = lane & 15;
  const int n0   = wave * 16;
  const float* seqb = seq + (size_t)b * S_ * HD_;

  v8f acc = {};
  const float rmax = rowmax[i];
  for (int kb = 0; kb < S_; kb += CH) {
    // fill chunk of exp(logit - rowmax), accumulate row sums
    float part = 0.f;
    for (int j = jt; j < CH; j += 8) {
      const int gj = kb + j;
      float l = leaky(fi + f1b[gj]) - 1e9f * (1.0f - adjrow[gj]);
      float p = expf(l - rmax);
      part += p;
      P[i][j] = p;
    }
    atomicAdd(&rowsum[i], part);
    __syncthreads();
    // WMMA: (16 x CH) @ (CH x 16) accumulate
#pragma unroll 4
    for (int kk = 0; kk < CH; kk += 4) {
      const int k  = kk + 2 * hi;
      const int gk = kb + k;
      v2f a = *(const v2f*)&P[ln][k];
      v2f bb; bb.x = seqb[(size_t)gk * HD_ + n0 + ln];
              bb.y = seqb[(size_t)(gk + 1) * HD_ + n0 + ln];
      acc = wmma4(a, bb, acc);
    }
    __syncthreads();
  }

  // normalize, add residual, elu, store industry[b, row, n]
#pragma unroll
  for (int r = 0; r < 8; ++r) {
    const int m  = r + 8 * hi;
    const int gr = row0 + m;
    float v = acc[r] / rowsum[m];
    v += resid[((size_t)b * S_ + gr) * HD_ + n0 + ln];
    out_ind[((size_t)b * S_ + gr) * HD_ + n0 + ln] = elu(v);
  }
}

// ---------------------------------------------------------------------------
// 4) intra-hyperedge attention: one block per (e,b); K=32 rows; fused
//    gather + GEMMs + softmax + scatter + maxpool.
// ---------------------------------------------------------------------------
__global__ __launch_bounds__(128) void intra_attn(
    const float* __restrict__ x, const int* __restrict__ member_idx,
    const float* __restrict__ W1, const float* __restrict__ b1,
    const float* __restrict__ a1p, const float* __restrict__ ba1p,
    const float* __restrict__ a2p, const float* __restrict__ ba2p,
    const float* __restrict__ Wr, const float* __restrict__ brp,
    float* __restrict__ out_all, float* __restrict__ out_he) {
  const int e = blockIdx.x / B_;
  const int b = blockIdx.x % B_;

  __shared__ float Xg[K_][F_];        // 16 KB gathered rows
  __shared__ float Sq[K_][HD_];       // 8 KB seq_fts
  __shared__ float P[K_][K_];         // 4 KB unnormalized coefs
  __shared__ float f1s[K_], f2s[K_], ssum[K_];
  __shared__ int   midx[K_];

  const int tid = threadIdx.x;
  if (tid < K_) midx[tid] = member_idx[e * K_ + tid];
  __syncthreads();

  // gather 32 rows x 128 floats with float4 loads
  for (int q = tid; q < K_ * (F_ / 4); q += 128) {
    const int r = q >> 5, c = q & 31;
    float4 v = ((const float4*)(x + ((size_t)b * S_ + midx[r]) * F_))[c];
    ((float4*)&Xg[r][0])[c] = v;
  }
  __syncthreads();

  const int wave = tid >> 5, lane = tid & 31;
  const int hi = lane >> 4, ln = lane & 15;
  const int n0 = wave * 16;
  const int n  = n0 + ln;

  // seq = Xg@W1 + b1 (kept in LDS), resid = Xg@Wr + br (kept in regs)
  v8f cs0 = {}, cs1 = {}, cr0 = {}, cr1 = {};
#pragma unroll
  for (int kk = 0; kk < F_; kk += 4) {
    const int k = kk + 2 * hi;
    v2f a0 = *(const v2f*)&Xg[ln][k];
    v2f a1v = *(const v2f*)&Xg[16 + ln][k];
    v2f bw; bw.x = W1[(size_t)k * HD_ + n]; bw.y = W1[(size_t)(k + 1) * HD_ + n];
    v2f bv; bv.x = Wr[(size_t)k * HD_ + n]; bv.y = Wr[(size_t)(k + 1) * HD_ + n];
    cs0 = wmma4(a0, bw, cs0);
    cs1 = wmma4(a1v, bw, cs1);
    cr0 = wmma4(a0, bv, cr0);
    cr1 = wmma4(a1v, bv, cr1);
  }
  const float bb1 = b1[n];
#pragma unroll
  for (int r = 0; r < 8; ++r) {
    Sq[r + 8 * hi][n]      = cs0[r] + bb1;
    Sq[16 + r + 8 * hi][n] = cs1[r] + bb1;
  }
  __syncthreads();

  // per-row f1/f2
  if (tid < K_) {
    float s1 = ba1p[0], s2 = ba2p[0];
#pragma unroll 8
    for (int k = 0; k < HD_; ++k) { float v = Sq[tid][k]; s1 += v * a1p[k]; s2 += v * a2p[k]; }
    f1s[tid] = s1; f2s[tid] = s2;
  }
  __syncthreads();

  // 32x32 logits -> unnormalized softmax in LDS (no mask for intra head)
  if (tid < K_) {
    const float fi = f2s[tid];
    float mx = -INFINITY;
#pragma unroll
    for (int j = 0; j < K_; ++j) mx = fmaxf(mx, leaky(fi + f1s[j]));
    float s = 0.f;
#pragma unroll
    for (int j = 0; j < K_; ++j) {
      float p = expf(leaky(fi + f1s[j]) - mx);
      s += p;
      P[tid][j] = p;
    }
    ssum[tid] = s;
  }
  __syncthreads();

  // ret = P @ Sq  (32x32 @ 32x64)
  v8f a0acc = {}, a1acc = {};
#pragma unroll
  for (int kk = 0; kk < K_; kk += 4) {
    const int k = kk + 2 * hi;
    v2f pa0 = *(const v2f*)&P[ln][k];
    v2f pa1 = *(const v2f*)&P[16 + ln][k];
    v2f bq; bq.x = Sq[k][n]; bq.y = Sq[k + 1][n];
    a0acc = wmma4(pa0, bq, a0acc);
    a1acc = wmma4(pa1, bq, a1acc);
  }

  // finalize: normalize, +resid, elu; fused scatter write and K-maxpool
  const float bbr = brp[n];
  float colmax = -INFINITY;
  const size_t base = ((size_t)e * B_ + b) * S_;
#pragma unroll
  for (int r = 0; r < 8; ++r) {
    const int m0 = r + 8 * hi, m1 = 16 + r + 8 * hi;
    float v0 = elu(a0acc[r] / ssum[m0] + cr0[r] + bbr);
    float v1 = elu(a1acc[r] / ssum[m1] + cr1[r] + bbr);
    out_all[(base + midx[m0]) * HD_ + n] = v0;
    out_all[(base + midx[m1]) * HD_ + n] = v1;
    colmax = fmaxf(colmax, fmaxf(v0, v1));
  }
  colmax = fmaxf(colmax, __shfl_xor(colmax, 16, 32));  // combine lane halves
  if (hi == 0) out_he[((size_t)b * E_ + e) * HD_ + n] = colmax;
}

// ---------------------------------------------------------------------------
extern "C" void kernel_launch(void* const* d_in, const int* in_sizes, int n_in,
                              void* d_out, int out_size, void* d_ws, size_t ws_size,
                              hipStream_t stream) {
  const float* x     = (const float*)d_in[0];
  const float* adj   = (const float*)d_in[1];
  const int*   midx  = (const int*)d_in[2];
  const float* W1_i  = (const float*)d_in[4];
  const float* b1_i  = (const float*)d_in[5];
  const float* a1_i  = (const float*)d_in[6];
  const float* ba1_i = (const float*)d_in[7];
  const float* a2_i  = (const float*)d_in[8];
  const float* ba2_i = (const float*)d_in[9];
  const float* Wr_i  = (const float*)d_in[10];
  const float* br_i  = (const float*)d_in[11];
  const float* W1_g  = (const float*)d_in[12];
  const float* b1_g  = (const float*)d_in[13];
  const float* a1_g  = (const float*)d_in[14];
  const float* ba1_g = (const float*)d_in[15];
  const float* a2_g  = (const float*)d_in[16];
  const float* ba2_g = (const float*)d_in[17];
  const float* Wr_g  = (const float*)d_in[18];
  const float* br_g  = (const float*)d_in[19];

  float* out_all = (float*)d_out;                               // [E,B,S,HD]
  float* out_he  = out_all + (size_t)E_ * B_ * S_ * HD_;        // [B,E,HD]
  float* out_ind = out_he + (size_t)B_ * E_ * HD_;              // [B,S,HD]

  float* seq   = (float*)d_ws;                                  // [B*S, HD]
  float* resid = seq + (size_t)B_ * S_ * HD_;                   // [B*S, HD]
  float* f1    = resid + (size_t)B_ * S_ * HD_;                 // [B*S]
  float* f2    = f1 + (size_t)B_ * S_;                          // [B*S]

  const size_t n4 = ((size_t)E_ * B_ * S_ * HD_) / 4;
  zero4<<<dim3((unsigned)(n4 / 256)), 256, 0, stream>>>((float4*)out_all, n4);

  prep_industry<<<dim3((B_ * S_) / 16), 128, 0, stream>>>(x, W1_g, b1_g, Wr_g,
                                                          br_g, seq, resid);
  f12_g<<<dim3((B_ * S_) / 256), 256, 0, stream>>>(seq, a1_g, ba1_g, a2_g, ba2_g,
                                                   f1, f2, B_ * S_);
  industry_attn<<<dim3(B_ * (S_ / 16)), 128, 0, stream>>>(seq, resid, f1, f2, adj,
                                                          out_ind);
  intra_attn<<<dim3(E_ * B_), 128, 0, stream>>>(x, midx, W1_i, b1_i, a1_i, ba1_i,
                                                a2_i, ba2_i, Wr_i, br_i, out_all,
                                                out_he);
}